// YOLOLoss_13125420057339
// MI455X (gfx1250) — compile-verified
//
#include <hip/hip_runtime.h>

// Problem constants (from reference)
#define NBATCH    4096
#define SGRID     14
#define CELLS     (NBATCH * SGRID * SGRID)   // 802816
#define CH        30                          // B*5 + C = 30
#define BLOCK     256
#define NBLOCKS   (CELLS / BLOCK)             // 3136 (exact)
#define CELL_SZ   (1.0f / 14.0f)

typedef __attribute__((ext_vector_type(2))) float v2f;
typedef __attribute__((ext_vector_type(8))) float v8f;

// Exact-f32 wave32 reduction using the CDNA5 matrix pipe:
// A (16x4 f32) = per-lane partials (K=0 for lanes 0-15, K=2 for lanes 16-31),
// B (4x16 f32) = ones  =>  D[m][n] = p(m) + p(m+16) for every n.
// Sum of the 8 D VGPRs gives rows 0-7 total (lanes 0-15) / rows 8-15 total
// (lanes 16-31); one shfl_xor(16) completes the wave total in all lanes.
__device__ __forceinline__ float wave_reduce_wmma(float p) {
    v2f a; a[0] = p;    a[1] = 0.0f;
    v2f b; b[0] = 1.0f; b[1] = 1.0f;
    v8f c = {};
    v8f d = __builtin_amdgcn_wmma_f32_16x16x4_f32(
        /*neg_a=*/false, a, /*neg_b=*/false, b,
        /*c_mod=*/(short)0, c, /*reuse_a=*/false, /*reuse_b=*/false);
    float s = d[0] + d[1] + d[2] + d[3] + d[4] + d[5] + d[6] + d[7];
    s += __shfl_xor(s, 16, 32);
    return s;
}

__global__ __launch_bounds__(BLOCK)
void yolo_loss_partial(const float* __restrict__ pred,
                       const float* __restrict__ tgt,
                       float* __restrict__ ws) {
    __shared__ __align__(16) float sp[8 * 32 * CH];  // 8 waves * 32 cells * 30 ch
    __shared__ __align__(16) float st[8 * 32 * CH];
    __shared__ float wpart[8];

    const int tid  = threadIdx.x;
    const int wid  = tid >> 5;
    const int lane = tid & 31;
    const int blockCell = blockIdx.x * BLOCK;
    const int waveCell  = blockCell + wid * 32;

    // ---- Stage 32 cells per wave: 960 floats = 480 v2f, fully coalesced, NT ----
    const v2f* gp = (const v2f*)(pred + (size_t)waveCell * CH);
    const v2f* gt = (const v2f*)(tgt  + (size_t)waveCell * CH);
    v2f* lp = (v2f*)(sp + wid * (32 * CH));
    v2f* lt = (v2f*)(st + wid * (32 * CH));
#pragma unroll
    for (int k = 0; k < 15; ++k) {
        lp[lane + k * 32] = __builtin_nontemporal_load(&gp[lane + k * 32]);
        lt[lane + k * 32] = __builtin_nontemporal_load(&gt[lane + k * 32]);
    }
    __syncthreads();

    const float* P = sp + wid * (32 * CH) + lane * CH;
    const float* T = st + wid * (32 * CH) + lane * CH;

    // ---- Per-cell loss ----
    const float t4 = T[4];
    float loss;
    if (t4 > 0.0f) {
        // target box = bt[0][:4]
        const float tminx = T[0] * CELL_SZ - 0.5f * T[2];
        const float tminy = T[1] * CELL_SZ - 0.5f * T[3];
        const float tmaxx = T[0] * CELL_SZ + 0.5f * T[2];
        const float tmaxy = T[1] * CELL_SZ + 0.5f * T[3];
        const float tarea = T[2] * T[3];

        float iou[2];
#pragma unroll
        for (int bb = 0; bb < 2; ++bb) {
            const float px = P[bb * 5 + 0], py = P[bb * 5 + 1];
            const float pw = P[bb * 5 + 2], ph = P[bb * 5 + 3];
            const float pminx = px * CELL_SZ - 0.5f * pw;
            const float pminy = py * CELL_SZ - 0.5f * ph;
            const float pmaxx = px * CELL_SZ + 0.5f * pw;
            const float pmaxy = py * CELL_SZ + 0.5f * ph;
            const float ix = fmaxf(fminf(pmaxx, tmaxx) - fmaxf(pminx, tminx), 0.0f);
            const float iy = fmaxf(fminf(pmaxy, tmaxy) - fmaxf(pminy, tminy), 0.0f);
            const float inter = ix * iy;
            iou[bb] = inter / (pw * ph + tarea - inter);
        }
        const int resp = (iou[1] > iou[0]) ? 1 : 0;      // jnp.argmax tie -> 0
        const float max_iou = fmaxf(iou[0], iou[1]);

        const float* pr = P + resp * 5;
        const float* tr = T + resp * 5;
        const float* pn = P + (1 - resp) * 5;

        float dc = pr[4] - max_iou;
        const float contain_l = dc * dc;
        const float notcon_l  = pn[4] * pn[4];

        const float dx = pr[0] - tr[0];
        const float dy = pr[1] - tr[1];
        const float dw = sqrtf(pr[2]) - sqrtf(tr[2]);
        const float dh = sqrtf(pr[3]) - sqrtf(tr[3]);
        const float loc_l = dx * dx + dy * dy + dw * dw + dh * dh;

        float cls_l = 0.0f;
#pragma unroll
        for (int cc = 0; cc < 20; ++cc) {
            const float dd = P[10 + cc] - T[10 + cc];
            cls_l += dd * dd;
        }
        loss = 5.0f * loc_l + 2.0f * contain_l + notcon_l + cls_l;
    } else {
        // noo term: both conf channels (target confs are 0 here)
        const float d4 = P[4] - t4;
        const float d9 = P[9] - T[9];
        loss = 0.5f * (d4 * d4 + d9 * d9);
    }

    // ---- Wave reduce (WMMA), then block reduce, one partial per block ----
    const float wsum = wave_reduce_wmma(loss);
    if (lane == 0) wpart[wid] = wsum;
    __syncthreads();
    if (tid == 0) {
        float bsum = 0.0f;
#pragma unroll
        for (int i = 0; i < 8; ++i) bsum += wpart[i];
        ws[blockIdx.x] = bsum;
    }
}

__global__ __launch_bounds__(256)
void yolo_loss_final(const float* __restrict__ ws, float* __restrict__ out) {
    __shared__ double red[256];
    const int tid = threadIdx.x;
    double acc = 0.0;
    for (int i = tid; i < NBLOCKS; i += 256) acc += (double)ws[i];
    red[tid] = acc;
    __syncthreads();
#pragma unroll
    for (int off = 128; off > 0; off >>= 1) {
        if (tid < off) red[tid] += red[tid + off];
        __syncthreads();
    }
    if (tid == 0) out[0] = (float)(red[0] / (double)NBATCH);
}

extern "C" void kernel_launch(void* const* d_in, const int* in_sizes, int n_in,
                              void* d_out, int out_size, void* d_ws, size_t ws_size,
                              hipStream_t stream) {
    const float* pred = (const float*)d_in[0];
    const float* tgt  = (const float*)d_in[1];
    float* out = (float*)d_out;
    float* ws  = (float*)d_ws;   // NBLOCKS floats of scratch, fully overwritten each call

    yolo_loss_partial<<<NBLOCKS, BLOCK, 0, stream>>>(pred, tgt, ws);
    yolo_loss_final<<<1, 256, 0, stream>>>(ws, out);
}